// GraphApp_81192061764218
// MI455X (gfx1250) — compile-verified
//
#include <hip/hip_runtime.h>

// GNN sum-aggregation + convex combine for MI455X (gfx1250, wave32).
//   out[n,d] = 0.9*(x[n,d] + sum_k agg[n,k,d]) + 0.1*h[n,d]
// Memory-bound (~0.92 GB traffic -> ~40us at 23.3 TB/s). The k-reduction is
// executed on the WMMA unit: V_WMMA_F32_16X16X4_F32 with A == ones performs a
// column sum of a 4x16 tile of agg (C[m,n] = sum_kk B[kk,n]); compute is free,
// so we use the matrix pipe as a reduction engine while keeping all global
// loads coalesced.

typedef __attribute__((ext_vector_type(2))) float v2f;
typedef __attribute__((ext_vector_type(8))) float v8f;

#define GNN_ALPHA 0.1f
#define K_NB 32
#define D_F  64

__global__ __launch_bounds__(256) void gnn_sum_wmma_kernel(
    const float* __restrict__ x,
    const float* __restrict__ agg,
    const float* __restrict__ h,
    float* __restrict__ out,
    int n_nodes)
{
    const int lane = threadIdx.x & 31;
    const int node = blockIdx.x * 8 + (threadIdx.x >> 5);   // one wave per node
    if (node >= n_nodes) return;                            // wave-uniform branch

    const int half = lane >> 4;   // 0: lanes 0-15, 1: lanes 16-31
    const int idx  = lane & 15;   // N position in B/C tiles -> d offset

    const float* __restrict__ base = agg + (size_t)node * (K_NB * D_F);

    // A = 16x4 ones (2 VGPRs/lane). Hoisted out of the loop by the compiler.
    v2f a; a.x = 1.0f; a.y = 1.0f;

    // Four C accumulators, one per 16-wide d-tile (d0 = 0,16,32,48).
    v8f c0 = {}, c1 = {}, c2 = {}, c3 = {};

    #pragma unroll
    for (int k0 = 0; k0 < K_NB; k0 += 4) {
        // B (4x16 f32): VGPR0 holds one K-row across lanes 0-15, another across
        // lanes 16-31; VGPR1 the next rows. Each load is 64B-contiguous per
        // half-wave (d is lane-contiguous). Any K<->slot permutation is fine:
        // A is all-ones, so C is the sum over the 4 loaded rows regardless.
        const float* r = base + (size_t)(k0 + 2 * half) * D_F + idx;
        v2f b0, b1, b2, b3;
        b0.x = r[0];   b0.y = r[D_F + 0];
        b1.x = r[16];  b1.y = r[D_F + 16];
        b2.x = r[32];  b2.y = r[D_F + 32];
        b3.x = r[48];  b3.y = r[D_F + 48];

        c0 = __builtin_amdgcn_wmma_f32_16x16x4_f32(false, a, false, b0, (short)0, c0, false, false);
        c1 = __builtin_amdgcn_wmma_f32_16x16x4_f32(false, a, false, b1, (short)0, c1, false, false);
        c2 = __builtin_amdgcn_wmma_f32_16x16x4_f32(false, a, false, b2, (short)0, c2, false, false);
        c3 = __builtin_amdgcn_wmma_f32_16x16x4_f32(false, a, false, b3, (short)0, c3, false, false);
    }

    // C rows are all identical (A == ones): lane l's element 0 of accumulator t
    // holds sum_k agg[node, :, 16*t + (l%16)]. Split the 4 d-tiles across the
    // two lane halves so all 32 lanes do the fused epilogue.
    float sa = (half == 0) ? c0[0] : c2[0];
    float sb = (half == 0) ? c1[0] : c3[0];
    const int d_a = half * 32 + idx;        // d0 = 0 or 32
    const int d_b = half * 32 + 16 + idx;   // d0 = 16 or 48

    const size_t o = (size_t)node * D_F;
    out[o + d_a] = (1.0f - GNN_ALPHA) * (x[o + d_a] + sa) + GNN_ALPHA * h[o + d_a];
    out[o + d_b] = (1.0f - GNN_ALPHA) * (x[o + d_b] + sb) + GNN_ALPHA * h[o + d_b];
}

// Bit-exact pass-through of the int32 `neighbor` array into the (float-typed)
// output buffer, vectorized to 16B transactions.
__global__ __launch_bounds__(256) void copy_i32x4_kernel(
    const int4* __restrict__ src, int4* __restrict__ dst, int n4)
{
    int i = blockIdx.x * blockDim.x + threadIdx.x;
    if (i < n4) dst[i] = src[i];
}

extern "C" void kernel_launch(void* const* d_in, const int* in_sizes, int n_in,
                              void* d_out, int out_size, void* d_ws, size_t ws_size,
                              hipStream_t stream) {
    const float* x   = (const float*)d_in[0];   // [N, D]
    const float* agg = (const float*)d_in[1];   // [N, K, D]
    const float* h   = (const float*)d_in[2];   // [N, D]
    const int*   nb  = (const int*)d_in[3];     // [N, K]
    float* out = (float*)d_out;

    const int n_nodes  = in_sizes[0] / D_F;     // 100000
    const int nb_count = in_sizes[3];           // N*K

    dim3 block(256);
    dim3 grid((n_nodes + 7) / 8);               // 8 waves/block, 1 node/wave
    hipLaunchKernelGGL(gnn_sum_wmma_kernel, grid, block, 0, stream,
                       x, agg, h, out, n_nodes);

    const int n4 = nb_count / 4;                // N*K divisible by 4
    dim3 grid2((n4 + 255) / 256);
    hipLaunchKernelGGL(copy_i32x4_kernel, grid2, block, 0, stream,
                       (const int4*)nb, (int4*)(out + (size_t)n_nodes * D_F), n4);
}